// GraphNetOriginal_9964324127505
// MI455X (gfx1250) — compile-verified
//
#include <hip/hip_runtime.h>
#include <math.h>

#define N_NODES 100000
#define N_EDGES 3200000
#define FEAT    32
#define NLAB    50
#define ELAB    10
#define OUTK    16

typedef __attribute__((ext_vector_type(2))) float v2f;
typedef __attribute__((ext_vector_type(8))) float v8f;

// ---------------------------------------------------------------- zero fill
__global__ void gnn_zero(float* __restrict__ p, int n) {
  int i = blockIdx.x * blockDim.x + threadIdx.x;
  if (i < n) p[i] = 0.0f;
}

// ------------------------------------------------- edge scatter (one wave32 per edge)
// lane = feature index. Edge metadata is wave-uniform -> readfirstlane to get
// scalar (SMEM) loads; x-row gather and atomic scatter are fully coalesced
// (one 128B line each). fp32 L2 atomics via unsafeAtomicAdd -> global_atomic_add_f32.
__global__ void gnn_edge_scatter(const float* __restrict__ x,
                                 const float* __restrict__ W,
                                 const int*   __restrict__ nl,
                                 const int*   __restrict__ esrc,
                                 const int*   __restrict__ edst,
                                 const int*   __restrict__ elab,
                                 float* __restrict__ agg) {
  int gid = blockIdx.x * blockDim.x + threadIdx.x;
  int e = gid >> 5;
  int f = gid & 31;
  if (e >= N_EDGES) return;
  int eu = __builtin_amdgcn_readfirstlane(e);

  // stream-prefetch edge metadata well ahead (edge lists are L2-resident after
  // layer 1, this keeps layer 1 fed) -> global_prefetch_b8
  if (f == 0) {
    int ep = eu + 4096;
    if (ep < N_EDGES) {
      __builtin_prefetch(&esrc[ep], 0, 0);
      __builtin_prefetch(&edst[ep], 0, 0);
      __builtin_prefetch(&elab[ep], 0, 0);
    }
  }

  int src = esrc[eu];
  int dst = edst[eu];
  int el  = elab[eu];
  int ls  = nl[src];
  int ld  = nl[dst];
  float w = W[(ls * NLAB + ld) * ELAB + el];

  float v = w * x[src * FEAT + f];
  unsafeAtomicAdd(&agg[dst * FEAT + f], v);
}

// ---------------------------------------------------------- bias + tanh (in place)
__global__ void gnn_bias_tanh(float* __restrict__ h,
                              const float* __restrict__ b,
                              const int*   __restrict__ nl) {
  int gid = blockIdx.x * blockDim.x + threadIdx.x;
  if (gid >= N_NODES * FEAT) return;
  int i = gid >> 5;               // node index (wave-uniform)
  h[gid] = tanhf(h[gid] + b[nl[i]]);
}

// ------------------------------------- per-label segment sum, LDS-blocked reduction
__global__ void gnn_label_sum(const float* __restrict__ h,
                              const int*   __restrict__ nl,
                              float* __restrict__ sums) {
  __shared__ float ls[NLAB * FEAT];          // 6400 B of the 320KB WGP LDS
  for (int j = threadIdx.x; j < NLAB * FEAT; j += blockDim.x) ls[j] = 0.0f;
  __syncthreads();

  int stride = gridDim.x * blockDim.x;
  for (int gid = blockIdx.x * blockDim.x + threadIdx.x;
       gid < N_NODES * FEAT; gid += stride) {
    int i = gid >> 5;
    int f = gid & 31;
    atomicAdd(&ls[nl[i] * FEAT + f], h[gid]);   // ds_add_f32
  }
  __syncthreads();

  for (int j = threadIdx.x; j < NLAB * FEAT; j += blockDim.x)
    unsafeAtomicAdd(&sums[j], ls[j]);           // one global atomic per entry per block
}

// ------------------------------------------- readout: (1x1600)x(1600x16) on WMMA f32
// D = A(16x4) * B(4x16) + C with V_WMMA_F32_16X16X4_F32, accumulating over 400 steps.
// A row M=0 carries the sums vector: lane0 holds K=0,1 (VGPR0,1), lane16 holds K=2,3.
// B: VGPR0 <-> K = 2*half, VGPR1 <-> K = 2*half+1, N = lane&15 (row-interleaved
// layout matching the documented A and C/D striping). Result row 0 lands in c[0]
// of lanes 0..15.
__global__ void gnn_resize_wmma(const float* __restrict__ sums,
                                const float* __restrict__ Wr,
                                const float* __restrict__ br,
                                float* __restrict__ out) {
  int lane = threadIdx.x;
  int n    = lane & 15;
  int half = lane >> 4;

  v8f c = {0.f, 0.f, 0.f, 0.f, 0.f, 0.f, 0.f, 0.f};

  for (int kk = 0; kk < NLAB * FEAT; kk += 4) {
    int kb = kk + 2 * half;

    v2f a; a.x = 0.0f; a.y = 0.0f;
    if (n == 0) {                       // lanes 0 and 16 carry row M=0 of A
      a.x = sums[kb];
      a.y = sums[kb + 1];
    }

    v2f bf;
    bf.x = Wr[kb * OUTK + n];
    bf.y = Wr[(kb + 1) * OUTK + n];

    c = __builtin_amdgcn_wmma_f32_16x16x4_f32(
        /*neg_a=*/false, a, /*neg_b=*/false, bf,
        /*c_mod=*/(short)0, c, /*reuse_a=*/false, /*reuse_b=*/false);
  }

  if (lane < 16) out[lane] = tanhf(c[0] + br[lane]);
}

// --------------------------------------------------------------------- launcher
extern "C" void kernel_launch(void* const* d_in, const int* in_sizes, int n_in,
                              void* d_out, int out_size, void* d_ws, size_t ws_size,
                              hipStream_t stream) {
  const float* x  = (const float*)d_in[0];
  const float* W1 = (const float*)d_in[1];
  const float* b1 = (const float*)d_in[2];
  const float* W2 = (const float*)d_in[3];
  const float* b2 = (const float*)d_in[4];
  const float* W3 = (const float*)d_in[5];
  const float* b3 = (const float*)d_in[6];
  const float* Wr = (const float*)d_in[7];
  const float* br = (const float*)d_in[8];
  const int* nl   = (const int*)d_in[9];
  const int* esrc = (const int*)d_in[10];
  const int* edst = (const int*)d_in[11];
  const int* elab = (const int*)d_in[12];
  float* out = (float*)d_out;

  const int NF = N_NODES * FEAT;               // 3.2M floats = 12.8 MB
  float* bufA = (float*)d_ws;
  float* bufB = bufA + NF;
  float* sums = bufB + NF;                     // 1600 floats

  const int zb = (NF + 255) / 256;
  const int eb = (N_EDGES * 32) / 256;         // exact: 400000 blocks
  const int tb = (NF + 255) / 256;

  // layer 1: x -> bufA
  gnn_zero<<<zb, 256, 0, stream>>>(bufA, NF);
  gnn_edge_scatter<<<eb, 256, 0, stream>>>(x,    W1, nl, esrc, edst, elab, bufA);
  gnn_bias_tanh<<<tb, 256, 0, stream>>>(bufA, b1, nl);

  // layer 2: bufA -> bufB
  gnn_zero<<<zb, 256, 0, stream>>>(bufB, NF);
  gnn_edge_scatter<<<eb, 256, 0, stream>>>(bufA, W2, nl, esrc, edst, elab, bufB);
  gnn_bias_tanh<<<tb, 256, 0, stream>>>(bufB, b2, nl);

  // layer 3: bufB -> bufA
  gnn_zero<<<zb, 256, 0, stream>>>(bufA, NF);
  gnn_edge_scatter<<<eb, 256, 0, stream>>>(bufB, W3, nl, esrc, edst, elab, bufA);
  gnn_bias_tanh<<<tb, 256, 0, stream>>>(bufA, b3, nl);

  // readout
  gnn_zero<<<(NLAB * FEAT + 255) / 256, 256, 0, stream>>>(sums, NLAB * FEAT);
  gnn_label_sum<<<512, 256, 0, stream>>>(bufA, nl, sums);
  gnn_resize_wmma<<<1, 32, 0, stream>>>(sums, Wr, br, out);
}